// freq_aug_decoder_21930103013982
// MI455X (gfx1250) — compile-verified
//
#include <hip/hip_runtime.h>
#include <math.h>

// ---------------------------------------------------------------------------
// Problem constants (from reference): B=64, C=512, S=512, E=257, LK=128, HK=129
// Output = irfft(attn(low,low,low) + attn(low,high,low))  (high_x is dead code)
// ---------------------------------------------------------------------------

typedef __attribute__((ext_vector_type(2))) float v2f;
typedef __attribute__((ext_vector_type(8))) float v8f;

#define TILE_BM 128   // block rows
#define TILE_BN 64    // block cols
#define TILE_KC 32    // K chunk staged in LDS
#define LDS_A_PITCH 34                 // even -> 8B-aligned (a0,a1) pairs, ds_load_b64
#define LDS_B_PITCH (2 * TILE_BN + 2)  // k-pair interleaved rows, even pitch

// ---------------------------------------------------------------------------
// Generic batched fp32 WMMA GEMM:
//   C[z] (accum? += : =) alpha * A[z] @ op(B[z])
// A: M x K row-major (lda), B: K x N row-major (ldb) or, if transB, N x K
// (so op(B)=B^T gives reduction over the stored row). C: M x N (ldc).
// 256 threads = 8 waves; wave w computes rows [w*16, w*16+16) x 64 cols.
//
// B chunk is stored k-pair interleaved in LDS:
//   element (kr, c)  ->  Bs[(kr>>1)*LDS_B_PITCH + 2*c + (kr&1)]
// so each lane's WMMA B fragment (B[k][col], B[k+1][col]) is one aligned b64.
// ---------------------------------------------------------------------------
__global__ void __launch_bounds__(256)
gemm_wmma_f32(const float* __restrict__ A, int lda, long long strideA,
              const float* __restrict__ B, int ldb, long long strideB,
              float* __restrict__ C, int ldc, long long strideC,
              int M, int N, int K, int transB, float alpha, int accum)
{
    __shared__ float As[TILE_BM * LDS_A_PITCH];
    __shared__ float Bs[(TILE_KC / 2) * LDS_B_PITCH];

    const int tid  = threadIdx.x;
    const int lane = tid & 31;
    const int half = lane >> 4;      // 0: lanes 0-15, 1: lanes 16-31
    const int lrow = lane & 15;
    const int wave = tid >> 5;       // 0..7

    const int m0 = blockIdx.y * TILE_BM;
    const int n0 = blockIdx.x * TILE_BN;

    A += (long long)blockIdx.z * strideA;
    B += (long long)blockIdx.z * strideB;
    C += (long long)blockIdx.z * strideC;

    v8f acc[4] = {};

    for (int kc = 0; kc < K; kc += TILE_KC) {
        // ---- stage A chunk (128 x 32), coalesced 32 floats per row ----
#pragma unroll
        for (int i = 0; i < (TILE_BM * TILE_KC) / 256; ++i) {
            int idx = i * 256 + tid;
            int r = idx >> 5, c = idx & (TILE_KC - 1);
            int gm = m0 + r, gk = kc + c;
            float v = (gm < M && gk < K) ? A[(long long)gm * lda + gk] : 0.0f;
            As[r * LDS_A_PITCH + c] = v;
        }
        // ---- stage B chunk (32 x 64), k-pair interleaved ----
#pragma unroll
        for (int i = 0; i < (TILE_KC * TILE_BN) / 256; ++i) {
            int idx = i * 256 + tid;
            float v;
            int kr, c;
            if (transB) {                       // B stored N x K: coalesce along K
                c  = idx >> 5;                  // 0..63
                kr = idx & 31;                  // 0..31
                int gn = n0 + c, gk = kc + kr;
                v = (gn < N && gk < K) ? B[(long long)gn * ldb + gk] : 0.0f;
            } else {                            // B stored K x N: coalesce along N
                kr = idx >> 6;                  // 0..31
                c  = idx & 63;                  // 0..63
                int gn = n0 + c, gk = kc + kr;
                v = (gk < K && gn < N) ? B[(long long)gk * ldb + gn] : 0.0f;
            }
            Bs[(kr >> 1) * LDS_B_PITCH + (c << 1) + (kr & 1)] = v;
        }
        __syncthreads();

        // ---- MMA over the chunk: 8 steps of K=4, 4 WMMAs per step ----
        const int arow = wave * 16 + lrow;      // this wave's A row in LDS
#pragma unroll
        for (int kk = 0; kk < TILE_KC; kk += 4) {
            // A fragment (16x4 f32): lanes 0-15 hold K=kk,kk+1; lanes 16-31 K=kk+2,kk+3
            // contiguous 8B-aligned pair -> single ds_load_b64
            const float* ap = &As[arow * LDS_A_PITCH + kk + 2 * half];
            v2f a;
            a[0] = ap[0];
            a[1] = ap[1];
            // B fragments (4x16 f32): pair row (kk/2 + half), contiguous b64 per lane
            const int prow = (kk >> 1) + half;
            v2f bf[4];
#pragma unroll
            for (int j = 0; j < 4; ++j) {
                const float* bp = &Bs[prow * LDS_B_PITCH + ((j * 16 + lrow) << 1)];
                bf[j][0] = bp[0];
                bf[j][1] = bp[1];
            }
#pragma unroll
            for (int j = 0; j < 4; ++j) {
                acc[j] = __builtin_amdgcn_wmma_f32_16x16x4_f32(
                    /*neg_a=*/false, a, /*neg_b=*/false, bf[j],
                    /*c_mod=*/(short)0, acc[j],
                    /*reuse_a=*/false, /*reuse_b=*/false);
            }
        }
        __syncthreads();
    }

    // ---- epilogue: C layout = 8 VGPRs, lanes 0-15 rows r, lanes 16-31 rows r+8
    const int wm0 = m0 + wave * 16;
#pragma unroll
    for (int j = 0; j < 4; ++j) {
        int col = n0 + j * 16 + lrow;
#pragma unroll
        for (int r = 0; r < 8; ++r) {
            int row = wm0 + r + 8 * half;
            if (row < M && col < N) {
                long long off = (long long)row * ldc + col;
                float v = alpha * acc[j][r];
                C[off] = accum ? (C[off] + v) : v;
            }
        }
    }
}

// ---------------------------------------------------------------------------
// DFT weight tables (ortho).  Forward (S x E):  Wcf[s,e]= cos(2pi s e/S)/sqrt(S),
// Wsf = -sin(.)/sqrt(S).  Inverse (E x S): Wci[e,s]= w_e cos(.)/sqrt(S),
// Wsi = -w_e sin(.)/sqrt(S), w_e = 1 for e in {0, S/2}, else 2 (irfft ignores
// imag of DC/Nyquist; sin term is exactly 0 there).  Argument reduced exactly
// via (s*e) mod S.
// ---------------------------------------------------------------------------
__global__ void gen_dft_weights(float* __restrict__ Wcf, float* __restrict__ Wsf,
                                float* __restrict__ Wci, float* __restrict__ Wsi,
                                int S, int E)
{
    int idx = blockIdx.x * blockDim.x + threadIdx.x;
    int total = S * E;
    if (idx >= total) return;
    const float invs = 1.0f / sqrtf((float)S);
    const float w0   = 6.28318530717958647692f / (float)S;

    // forward layout: idx -> (s, e)
    int s = idx / E, e = idx - s * E;
    int t = (s * e) & (S - 1);
    float sn, cs;
    sincosf((float)t * w0, &sn, &cs);
    Wcf[idx] =  cs * invs;
    Wsf[idx] = -sn * invs;

    // inverse layout: idx -> (e2, s2); total == E*S as well
    int e2 = idx / S, s2 = idx - e2 * S;
    int t2 = (e2 * s2) & (S - 1);
    float sn2, cs2;
    sincosf((float)t2 * w0, &sn2, &cs2);
    float sc = (e2 == 0 || e2 == (S >> 1)) ? 1.0f : 2.0f;
    Wci[idx] =  sc * cs2 * invs;
    Wsi[idx] = -sc * sn2 * invs;
}

// ---------------------------------------------------------------------------
// Complex bias + CReLU, in place on separate real/imag planes (rows x cols).
// ---------------------------------------------------------------------------
__global__ void bias_crelu(float* __restrict__ R, float* __restrict__ I,
                           const float* __restrict__ br, const float* __restrict__ bi,
                           int total, int cols)
{
    int idx = blockIdx.x * blockDim.x + threadIdx.x;
    if (idx >= total) return;
    int c = idx % cols;
    R[idx] = fmaxf(R[idx] + br[c], 0.0f);
    I[idx] = fmaxf(I[idx] + bi[c], 0.0f);
}

// ---------------------------------------------------------------------------
// Row softmax in place, one 256-thread block per row (cols = 512 here).
// ---------------------------------------------------------------------------
__global__ void __launch_bounds__(256)
softmax_rows(float* __restrict__ P, int cols)
{
    __shared__ float red[256];
    float* p = P + (long long)blockIdx.x * cols;
    const int t = threadIdx.x;

    float m = -INFINITY;
    for (int c = t; c < cols; c += 256) m = fmaxf(m, p[c]);
    red[t] = m;
    __syncthreads();
    for (int s = 128; s > 0; s >>= 1) {
        if (t < s) red[t] = fmaxf(red[t], red[t + s]);
        __syncthreads();
    }
    m = red[0];
    __syncthreads();

    float sum = 0.0f;
    for (int c = t; c < cols; c += 256) {
        float e = expf(p[c] - m);
        p[c] = e;
        sum += e;
    }
    red[t] = sum;
    __syncthreads();
    for (int s = 128; s > 0; s >>= 1) {
        if (t < s) red[t] += red[t + s];
        __syncthreads();
    }
    float inv = 1.0f / red[0];
    for (int c = t; c < cols; c += 256) p[c] *= inv;
}

// ---------------------------------------------------------------------------
// Host-side GEMM launcher
// ---------------------------------------------------------------------------
static inline void launch_gemm(hipStream_t st,
                               const float* A, int lda, long long sA,
                               const float* B, int ldb, long long sB,
                               float* C, int ldc, long long sC,
                               int M, int N, int K, int batch,
                               int transB, float alpha, int accum)
{
    dim3 grid((N + TILE_BN - 1) / TILE_BN, (M + TILE_BM - 1) / TILE_BM, batch);
    gemm_wmma_f32<<<grid, dim3(256), 0, st>>>(A, lda, sA, B, ldb, sB, C, ldc, sC,
                                              M, N, K, transB, alpha, accum);
}

extern "C" void kernel_launch(void* const* d_in, const int* in_sizes, int n_in,
                              void* d_out, int out_size, void* d_ws, size_t ws_size,
                              hipStream_t stream)
{
    (void)in_sizes; (void)n_in; (void)out_size; (void)ws_size;

    const int Bz = 64, Cc = 512, Sn = 512, En = 257, LK = 128, HK = 129;
    const int BC = Bz * Cc;                       // 32768 rows
    const float scale = 1.0f / sqrtf((float)En);  // attention scale (pre-softmax)

    const float* x    = (const float*)d_in[0];
    const float* l1r  = (const float*)d_in[1];
    const float* l1i  = (const float*)d_in[2];
    const float* h1r  = (const float*)d_in[3];
    const float* h1i  = (const float*)d_in[4];
    const float* lb1r = (const float*)d_in[5];
    const float* lb1i = (const float*)d_in[6];
    const float* hb1r = (const float*)d_in[7];
    const float* hb1i = (const float*)d_in[8];
    float* out = (float*)d_out;

    // ---- workspace bump allocator (floats) ----
    float* ws = (float*)d_ws;
    size_t off = 0;
    auto alloc = [&](size_t n) { float* p = ws + off; off += n; return p; };

    const size_t WSZ = (size_t)Sn * En;           // 131584
    const size_t XSZ = (size_t)BC * En;           // 8,421,376
    const size_t SSZ = (size_t)Bz * Cc * Cc;      // 16,777,216

    float* Wcf = alloc(WSZ);  float* Wsf = alloc(WSZ);
    float* Wci = alloc(WSZ);  float* Wsi = alloc(WSZ);
    float* XR  = alloc(XSZ);  float* XI  = alloc(XSZ);
    float* LR  = alloc(XSZ);  float* LI  = alloc(XSZ);
    float* HR  = alloc(XSZ);  float* HI  = alloc(XSZ);
    float* AR  = alloc(XSZ);  float* AI  = alloc(XSZ);
    float* SR  = alloc(SSZ);  float* SI  = alloc(SSZ);

    // ---- 0. DFT weight tables ----
    {
        int tot = (int)WSZ;
        gen_dft_weights<<<(tot + 255) / 256, 256, 0, stream>>>(Wcf, Wsf, Wci, Wsi, Sn, En);
    }

    // ---- 1. rfft(x, ortho) as GEMM: X = x @ W  (32768x512 @ 512x257) ----
    launch_gemm(stream, x, Sn, 0, Wcf, En, 0, XR, En, 0, BC, En, Sn, 1, 0, 1.0f, 0);
    launch_gemm(stream, x, Sn, 0, Wsf, En, 0, XI, En, 0, BC, En, Sn, 1, 0, 1.0f, 0);

    // ---- 2. spectral mixing (complex matmul = 4 real GEMMs) + bias + CReLU ----
    // low: uses freqs [0,128), weights (128 x 257)
    launch_gemm(stream, XR, En, 0, l1r, En, 0, LR, En, 0, BC, En, LK, 1, 0,  1.0f, 0);
    launch_gemm(stream, XI, En, 0, l1i, En, 0, LR, En, 0, BC, En, LK, 1, 0, -1.0f, 1);
    launch_gemm(stream, XR, En, 0, l1i, En, 0, LI, En, 0, BC, En, LK, 1, 0,  1.0f, 0);
    launch_gemm(stream, XI, En, 0, l1r, En, 0, LI, En, 0, BC, En, LK, 1, 0,  1.0f, 1);
    bias_crelu<<<((int)XSZ + 255) / 256, 256, 0, stream>>>(LR, LI, lb1r, lb1i, (int)XSZ, En);

    // high: uses freqs [128,257), weights (129 x 257) — A offset +128 within rows
    launch_gemm(stream, XR + LK, En, 0, h1r, En, 0, HR, En, 0, BC, En, HK, 1, 0,  1.0f, 0);
    launch_gemm(stream, XI + LK, En, 0, h1i, En, 0, HR, En, 0, BC, En, HK, 1, 0, -1.0f, 1);
    launch_gemm(stream, XR + LK, En, 0, h1i, En, 0, HI, En, 0, BC, En, HK, 1, 0,  1.0f, 0);
    launch_gemm(stream, XI + LK, En, 0, h1r, En, 0, HI, En, 0, BC, En, HK, 1, 0,  1.0f, 1);
    bias_crelu<<<((int)XSZ + 255) / 256, 256, 0, stream>>>(HR, HI, hb1r, hb1i, (int)XSZ, En);

    // ---- 3. live attentions only: A += softmax_c(scale*low@key^T) @ low ----
    const long long sL = (long long)Cc * En;      // per-batch stride of low/high
    const long long sS = (long long)Cc * Cc;      // per-batch stride of scores
    for (int pass = 0; pass < 2; ++pass) {
        const float* KRm = (pass == 0) ? LR : HR;
        const float* KIm = (pass == 0) ? LI : HI;
        const int first = (pass == 0);
        // scores = scale * (QR@KR^T - QI@KI^T) + i * scale * (QR@KI^T + QI@KR^T)
        launch_gemm(stream, LR, En, sL, KRm, En, sL, SR, Cc, sS, Cc, Cc, En, Bz, 1,  scale, 0);
        launch_gemm(stream, LI, En, sL, KIm, En, sL, SR, Cc, sS, Cc, Cc, En, Bz, 1, -scale, 1);
        launch_gemm(stream, LR, En, sL, KIm, En, sL, SI, Cc, sS, Cc, Cc, En, Bz, 1,  scale, 0);
        launch_gemm(stream, LI, En, sL, KRm, En, sL, SI, Cc, sS, Cc, Cc, En, Bz, 1,  scale, 1);
        // csoftmax: independent row softmax on real and imag planes
        softmax_rows<<<Bz * Cc, 256, 0, stream>>>(SR, Cc);
        softmax_rows<<<Bz * Cc, 256, 0, stream>>>(SI, Cc);
        // P @ V, V = low (complex): AR (+)= PR@LR - PI@LI ; AI (+)= PR@LI + PI@LR
        launch_gemm(stream, SR, Cc, sS, LR, En, sL, AR, En, sL, Cc, En, Cc, Bz, 0,  1.0f, first ? 0 : 1);
        launch_gemm(stream, SI, Cc, sS, LI, En, sL, AR, En, sL, Cc, En, Cc, Bz, 0, -1.0f, 1);
        launch_gemm(stream, SR, Cc, sS, LI, En, sL, AI, En, sL, Cc, En, Cc, Bz, 0,  1.0f, first ? 0 : 1);
        launch_gemm(stream, SI, Cc, sS, LR, En, sL, AI, En, sL, Cc, En, Cc, Bz, 0,  1.0f, 1);
    }

    // ---- 4. irfft(low_x, n=512, ortho) as GEMM: out = AR@Wci + AI@Wsi ----
    launch_gemm(stream, AR, En, 0, Wci, Sn, 0, out, Sn, 0, BC, Sn, En, 1, 0, 1.0f, 0);
    launch_gemm(stream, AI, En, 0, Wsi, Sn, 0, out, Sn, 0, BC, Sn, En, 1, 0, 1.0f, 1);
}